// STDABottleneck_77790447665636
// MI455X (gfx1250) — compile-verified
//
#include <hip/hip_runtime.h>
#include <math.h>

typedef __attribute__((ext_vector_type(2))) float v2f;
typedef __attribute__((ext_vector_type(8))) float v8f;

#define BB 8
#define CC 64
#define HH 128
#define WW 128
#define HWHW (HH * WW)

// ---------------------------------------------------------------------------
// Pack OIHW 3x3 weights into [tap][k/2][n][2] so a lane's B-fragment for the
// f32 16x16x4 WMMA (K pair 2*half, 2*half+1 at column n) is one contiguous,
// coalesced float2 load.
// ---------------------------------------------------------------------------
__global__ __launch_bounds__(256) void pack_weights_k(const float* __restrict__ Wsrc,
                                                      float* __restrict__ Wp,
                                                      int NOUT) {
  int idx = blockIdx.x * 256 + threadIdx.x;
  int total = NOUT * CC * 9;
  if (idx >= total) return;
  int tap = idx % 9;        // ky*3 + kx
  int t   = idx / 9;
  int c   = t % CC;         // input channel
  int o   = t / CC;         // output channel
  float v = Wsrc[idx];      // flat OIHW: ((o*CC + c)*3 + ky)*3 + kx
  Wp[((tap * (CC / 2) + (c >> 1)) * NOUT + o) * 2 + (c & 1)] = v;
}

// ---------------------------------------------------------------------------
// 3x3 same-pad conv as implicit GEMM on V_WMMA_F32_16X16X4_F32.
//   M tile  : 64 px of one image row (4 sub-tiles of 16, one per wave group)
//   N tile  : all NOUT channels (2 wave groups over N, NPW sub-tiles each)
//   K order : tap-major, 64 input channels inner, stepped by 4
// DEFORM=true fuses the clamped 4-corner bilinear gather into LDS staging
// (main conv) and adds the bias; DEFORM=false is the plain offset conv.
// ---------------------------------------------------------------------------
template <int NOUT, bool DEFORM>
__global__ __launch_bounds__(256) void conv3x3_wmma_k(
    const float* __restrict__ x, const float* __restrict__ off,
    const float* __restrict__ Wp, const float* __restrict__ bias,
    float* __restrict__ out) {
  // [3 rows][66 px][64 ch padded to 68] -> conflict-free float2 A reads.
  __shared__ float lds[3 * 66 * 68];

  const int tid = threadIdx.x;
  const int w0  = blockIdx.x * 64;
  const int h   = blockIdx.y;
  const int b   = blockIdx.z;

  // ---- Stage input halo tile (with fused deformable gather if DEFORM) ----
  for (int f = tid; f < 3 * 66 * CC; f += 256) {
    int px  = f % 66;            // px fastest -> coalesced global reads
    int t   = f / 66;
    int ch  = t % CC;
    int row = t / CC;
    int gy  = h + row - 1;
    int gx  = w0 + px - 1;
    float v = 0.0f;
    if (gy >= 0 && gy < HH && gx >= 0 && gx < WW) {
      if (DEFORM) {
        int bc = b * CC + ch;
        int p  = gy * WW + gx;
        const float* ob = off + (size_t)bc * (2 * HWHW);
        float dy = ob[2 * p];
        float dx = ob[2 * p + 1];
        float cy = fminf(fmaxf((float)gy + dy, 0.0f), (float)(HH - 1));
        float cx = fminf(fmaxf((float)gx + dx, 0.0f), (float)(WW - 1));
        float fy0 = floorf(cy), fx0 = floorf(cx);
        float fy1 = ceilf(cy),  fx1 = ceilf(cx);
        int y0 = (int)fy0, x0i = (int)fx0, y1 = (int)fy1, x1i = (int)fx1;
        const float* xb = x + (size_t)bc * HWHW;
        float v_lt = xb[y0 * WW + x0i];
        float v_rb = xb[y1 * WW + x1i];
        float v_lb = xb[y0 * WW + x1i];
        float v_rt = xb[y1 * WW + x0i];
        float wy = cy - fy0, wx = cx - fx0;
        float vt  = wy * (v_rt - v_lt) + v_lt;
        float vbt = wy * (v_rb - v_lb) + v_lb;
        v = wx * (vbt - vt) + vt;
      } else {
        v = x[((size_t)(b * CC + ch) * HH + gy) * WW + gx];
      }
    }
    lds[(row * 66 + px) * 68 + ch] = v;
  }
  __syncthreads();

  constexpr int NPW = NOUT / 32;  // N sub-tiles per wave
  const int wave = tid >> 5;
  const int lane = tid & 31;
  const int msub = wave & 3;      // which 16-px M sub-tile
  const int ngrp = wave >> 2;     // which half of N
  const int m    = lane & 15;     // row (A) / col (B,D) within sub-tile
  const int kh   = lane >> 4;     // K half: lanes 0-15 -> K{0,1}, 16-31 -> K{2,3}

  v8f acc[NPW];
#pragma unroll
  for (int j = 0; j < NPW; ++j) acc[j] = v8f{};

  for (int tap = 0; tap < 9; ++tap) {
    const int ky = tap / 3;
    const int kx = tap % 3;
    const float* wtap = Wp + (size_t)tap * (CC / 2) * NOUT * 2;
    if (tap < 8)  // warm L2/WGP$ for the next tap's weight slab
      __builtin_prefetch(Wp + (size_t)(tap + 1) * (CC / 2) * NOUT * 2, 0, 1);
    const int arow = (ky * 66 + msub * 16 + m + kx) * 68;
#pragma unroll 2
    for (int kc = 0; kc < CC; kc += 4) {
      v2f a = *(const v2f*)&lds[arow + kc + 2 * kh];
#pragma unroll
      for (int j = 0; j < NPW; ++j) {
        int n = (ngrp * NPW + j) * 16 + m;
        v2f bf = *(const v2f*)&wtap[(((kc >> 1) + kh) * NOUT + n) * 2];
        // D = A(16x4 f32) * B(4x16 f32) + C(16x16 f32)
        acc[j] = __builtin_amdgcn_wmma_f32_16x16x4_f32(
            false, a, false, bf, (short)0, acc[j], false, false);
      }
    }
  }

  __syncthreads();  // input tile dead; reuse LDS for the store transpose

  // D layout: lane -> col n, vgpr r -> row r + 8*(lane/16)
#pragma unroll
  for (int j = 0; j < NPW; ++j) {
    int n = (ngrp * NPW + j) * 16 + m;
#pragma unroll
    for (int r = 0; r < 8; ++r) {
      int px = msub * 16 + r + 8 * kh;
      lds[n * 66 + px] = acc[j][r];
    }
  }
  __syncthreads();

  for (int f = tid; f < NOUT * 64; f += 256) {
    int n  = f >> 6;
    int px = f & 63;            // px fastest -> coalesced global stores
    float v = lds[n * 66 + px];
    if (DEFORM) v += bias[n];
    out[(((size_t)b * NOUT + n) * HH + h) * WW + w0 + px] = v;
  }
}

// ---------------------------------------------------------------------------
extern "C" void kernel_launch(void* const* d_in, const int* in_sizes, int n_in,
                              void* d_out, int out_size, void* d_ws, size_t ws_size,
                              hipStream_t stream) {
  const float* x      = (const float*)d_in[0];   // (8,64,128,128)
  const float* W_off  = (const float*)d_in[1];   // (128,64,3,3)
  const float* W_conv = (const float*)d_in[2];   // (64,64,3,3)
  const float* b_conv = (const float*)d_in[3];   // (64,)
  float* out = (float*)d_out;                    // (8,64,128,128)

  char* ws = (char*)d_ws;
  float* offsets = (float*)ws;  // (8,128,128,128) fp32 = 67.1 MB
  size_t offBytes = (size_t)BB * 2 * CC * HH * WW * sizeof(float);
  float* WpOff  = (float*)(ws + offBytes);       // packed (128,64,3,3)
  float* WpConv = WpOff + 2 * CC * CC * 9;       // packed (64,64,3,3)

  pack_weights_k<<<(2 * CC * CC * 9 + 255) / 256, 256, 0, stream>>>(W_off, WpOff, 2 * CC);
  pack_weights_k<<<(CC * CC * 9 + 255) / 256, 256, 0, stream>>>(W_conv, WpConv, CC);

  dim3 grid(WW / 64, HH, BB);  // 2 x 128 x 8 = 2048 blocks, 8 waves each
  // 1) offset conv: x -> offsets (N=128)
  conv3x3_wmma_k<2 * CC, false><<<grid, 256, 0, stream>>>(x, nullptr, WpOff, nullptr, offsets);
  // 2) fused deform-gather + main conv + bias: -> out (N=64)
  conv3x3_wmma_k<CC, true><<<grid, 256, 0, stream>>>(x, offsets, WpConv, b_conv, out);
}